// E_GCL_36524401885239
// MI455X (gfx1250) — compile-verified
//
#include <hip/hip_runtime.h>

typedef float v2f __attribute__((ext_vector_type(2)));
typedef float v8f __attribute__((ext_vector_type(8)));

#define D 128
#define A_LD 260   // 256 + 4 pad: LDS banks (4*m)%64 distinct -> conflict-free b64 reads
#define E_LD 132   // 128 + 4 pad
#define TM 32      // M rows (edges/nodes) per block tile
#define NT 8       // n-tiles per block (one per wave32)
#define KS1 64     // K-steps for K=256 GEMM (step 4)
#define KS2 32     // K-steps for K=128 GEMM (step 4)

__device__ __forceinline__ float silu_f(float x) {
  return x / (1.0f + __expf(-x));
}

// D = A*B + C, 16x16 tile, K=4, full fp32 (V_WMMA_F32_16X16X4_F32)
__device__ __forceinline__ v8f wmma_k4(v2f a, v2f b, v8f c) {
  return __builtin_amdgcn_wmma_f32_16x16x4_f32(false, a, false, b, (short)0, c,
                                               false, false);
}

// Stage W[K x 128] (row-major) into LDS, pre-swizzled to the WMMA B-operand
// per-lane layout: Bs[(nt*nks + ks)*32 + lane] = { W[4ks+2h][16nt+m],
//                                                  W[4ks+2h+1][16nt+m] }
__device__ __forceinline__ void stage_B(const float* __restrict__ W,
                                        v2f* __restrict__ Bs, int nks) {
  const int total = NT * nks * 32;
  for (int idx = threadIdx.x; idx < total; idx += blockDim.x) {
    const int lane = idx & 31;
    const int ks = (idx >> 5) % nks;
    const int nt = (idx >> 5) / nks;
    const int m = lane & 15;
    const int hh = lane >> 4;
    const int k = 4 * ks + 2 * hh;
    const int n = 16 * nt + m;
    v2f w;
    w.x = W[k * D + n];
    w.y = W[(k + 1) * D + n];
    Bs[idx] = w;
  }
}

__global__ __launch_bounds__(256) void zero_f32(float* __restrict__ p, long n) {
  long i = (long)blockIdx.x * blockDim.x + threadIdx.x;
  if (i < n) p[i] = 0.0f;
}

__global__ __launch_bounds__(256) void coord_mod_kernel(
    const float* __restrict__ coord, const float* __restrict__ l,
    float* __restrict__ out, int n3) {
  int i = blockIdx.x * blockDim.x + threadIdx.x;
  if (i < n3) {
    float L = l[i % 3];
    float r = fmodf(coord[i], L);
    if (r < 0.0f) r += L;  // jnp.remainder: sign of divisor
    out[i] = r;
  }
}

// ---------------- edge model: silu(silu([h_r|h_c|rad]@We1+be1)@We2+be2) -----
__global__ __launch_bounds__(256) void egcl_edge_kernel(
    const float* __restrict__ h, const int* __restrict__ ei,
    const float* __restrict__ coord, const float* __restrict__ l,
    const float* __restrict__ We1, const float* __restrict__ be1,
    const float* __restrict__ We2, const float* __restrict__ be2,
    float* __restrict__ edge_feat, float* __restrict__ agg, int E) {
  __shared__ v2f Bs1[NT * KS1 * 32];  // 128 KB: We1 rows 0..255, swizzled
  __shared__ v2f Bs2[NT * KS2 * 32];  // 64 KB: We2, swizzled
  __shared__ float As[TM * A_LD];     // 33.3 KB
  __shared__ float Es[TM * E_LD];     // 16.9 KB
  __shared__ float rad[TM];
  __shared__ int ridx[TM];

  stage_B(We1, Bs1, KS1);
  stage_B(We2, Bs2, KS2);

  const int t = threadIdx.x;
  const int lane = t & 31;
  const int wv = t >> 5;  // wave -> N-tile
  const int m = lane & 15;
  const int hf = lane >> 4;
  const int h2 = hf * 2;
  const int n0 = wv * 16;

  const float w256 = We1[(long)(2 * D) * D + n0 + m];  // radial row of We1
  const float b1 = be1[n0 + m];
  const float b2 = be2[n0 + m];

  const long ntiles = ((long)E + TM - 1) / TM;
  for (long tile = blockIdx.x; tile < ntiles; tile += gridDim.x) {
    const long base = tile * TM;
    __syncthreads();  // staging done / previous iteration's reads done

    {  // gather h[row], h[col] -> LDS A-matrix [32][256]; 8 threads per edge
      const int el = t >> 3;  // edge slot 0..31
      const int ch = t & 7;   // 8 chunks of 32 floats
      long e = base + el;
      const long es = (e < E) ? e : 0;
      const int r = ei[es];
      const int c = ei[E + es];
      const float4* src =
          (const float4*)(h + (long)(ch < 4 ? r : c) * D) + (ch & 3) * 8;
      float4* dst = (float4*)(As + el * A_LD + ch * 32);
#pragma unroll
      for (int j = 0; j < 8; ++j) dst[j] = src[j];
      if (ch == 0) {  // radial with minimum-image wrap
        float s = 0.0f;
#pragma unroll
        for (int d = 0; d < 3; ++d) {
          float L = l[d];
          float df = coord[(long)r * 3 + d] - coord[(long)c * 3 + d];
          df = (df > 0.5f * L) ? df - L : df;
          df = (df < -0.5f * L) ? df + L : df;
          s += df * df;
        }
        rad[el] = s;
        ridx[el] = r;
      }
    }
    __syncthreads();

    // GEMM1: [32,256] x [256,128]-tile; two M-tiles share each B fetch.
    v8f acc0 = {0.f, 0.f, 0.f, 0.f, 0.f, 0.f, 0.f, 0.f};
    v8f acc1 = {0.f, 0.f, 0.f, 0.f, 0.f, 0.f, 0.f, 0.f};
    {
      const float* arow = As + m * A_LD + h2;
      const v2f* bp = Bs1 + wv * (KS1 * 32) + lane;
#pragma unroll
      for (int ks = 0; ks < KS1; ++ks) {
        v2f b = bp[ks * 32];
        v2f a0 = *(const v2f*)(arow + 4 * ks);
        v2f a1 = *(const v2f*)(arow + 16 * A_LD + 4 * ks);
        acc0 = wmma_k4(a0, b, acc0);
        acc1 = wmma_k4(a1, b, acc1);
      }
    }
    // rank-1 radial term + bias + SiLU -> Es (GEMM2 A-matrix)
#pragma unroll
    for (int rr = 0; rr < 8; ++rr) {
      const int M = rr + 8 * hf;
      float x0 = acc0[rr] + rad[M] * w256 + b1;
      float x1 = acc1[rr] + rad[16 + M] * w256 + b1;
      Es[M * E_LD + n0 + m] = silu_f(x0);
      Es[(16 + M) * E_LD + n0 + m] = silu_f(x1);
    }
    __syncthreads();

    // GEMM2: [32,128] x [128,128]-tile
    v8f d0 = {0.f, 0.f, 0.f, 0.f, 0.f, 0.f, 0.f, 0.f};
    v8f d1 = {0.f, 0.f, 0.f, 0.f, 0.f, 0.f, 0.f, 0.f};
    {
      const float* arow = Es + m * E_LD + h2;
      const v2f* bp = Bs2 + wv * (KS2 * 32) + lane;
#pragma unroll
      for (int ks = 0; ks < KS2; ++ks) {
        v2f b = bp[ks * 32];
        v2f a0 = *(const v2f*)(arow + 4 * ks);
        v2f a1 = *(const v2f*)(arow + 16 * E_LD + 4 * ks);
        d0 = wmma_k4(a0, b, d0);
        d1 = wmma_k4(a1, b, d1);
      }
    }
#pragma unroll
    for (int rr = 0; rr < 8; ++rr) {
      const int M0 = rr + 8 * hf;
#pragma unroll
      for (int half = 0; half < 2; ++half) {
        const int M = M0 + 16 * half;
        const long e = base + M;
        if (e < E) {
          float v = half ? d1[rr] : d0[rr];
          float s = silu_f(v + b2);
          edge_feat[e * D + n0 + m] = s;
          __hip_atomic_fetch_add(&agg[(long)ridx[M] * D + n0 + m], s,
                                 __ATOMIC_RELAXED, __HIP_MEMORY_SCOPE_AGENT);
        }
      }
    }
  }
}

// ---------------- node model: h + silu([h|agg]@Wn1+bn1)@Wn2 + bn2 ----------
__global__ __launch_bounds__(256) void egcl_node_kernel(
    const float* __restrict__ h, const float* __restrict__ agg,
    const float* __restrict__ Wn1, const float* __restrict__ bn1,
    const float* __restrict__ Wn2, const float* __restrict__ bn2,
    float* __restrict__ h_out, int N) {
  __shared__ v2f Bs1[NT * KS1 * 32];  // 128 KB: Wn1 swizzled
  __shared__ v2f Bs2[NT * KS2 * 32];  // 64 KB: Wn2 swizzled
  __shared__ float As[TM * A_LD];
  __shared__ float Ns[TM * E_LD];

  stage_B(Wn1, Bs1, KS1);
  stage_B(Wn2, Bs2, KS2);

  const int t = threadIdx.x;
  const int lane = t & 31;
  const int wv = t >> 5;
  const int m = lane & 15;
  const int hf = lane >> 4;
  const int h2 = hf * 2;
  const int n0 = wv * 16;

  const float b1 = bn1[n0 + m];
  const float b2 = bn2[n0 + m];

  const long ntiles = ((long)N + TM - 1) / TM;
  for (long tile = blockIdx.x; tile < ntiles; tile += gridDim.x) {
    const long base = tile * TM;
    __syncthreads();

    {  // gather [h | agg] -> LDS A-matrix [32][256]
      const int nl = t >> 3;
      const int ch = t & 7;
      long g = base + nl;
      const long gs = (g < N) ? g : 0;
      const float* srow = (ch < 4) ? (h + gs * D) : (agg + gs * D);
      const float4* src = (const float4*)srow + (ch & 3) * 8;
      float4* dst = (float4*)(As + nl * A_LD + ch * 32);
#pragma unroll
      for (int j = 0; j < 8; ++j) dst[j] = src[j];
    }
    __syncthreads();

    // GEMM1: [32,256] x [256,128]-tile
    v8f acc0 = {0.f, 0.f, 0.f, 0.f, 0.f, 0.f, 0.f, 0.f};
    v8f acc1 = {0.f, 0.f, 0.f, 0.f, 0.f, 0.f, 0.f, 0.f};
    {
      const float* arow = As + m * A_LD + h2;
      const v2f* bp = Bs1 + wv * (KS1 * 32) + lane;
#pragma unroll
      for (int ks = 0; ks < KS1; ++ks) {
        v2f b = bp[ks * 32];
        v2f a0 = *(const v2f*)(arow + 4 * ks);
        v2f a1 = *(const v2f*)(arow + 16 * A_LD + 4 * ks);
        acc0 = wmma_k4(a0, b, acc0);
        acc1 = wmma_k4(a1, b, acc1);
      }
    }
#pragma unroll
    for (int rr = 0; rr < 8; ++rr) {
      const int M = rr + 8 * hf;
      Ns[M * E_LD + n0 + m] = silu_f(acc0[rr] + b1);
      Ns[(16 + M) * E_LD + n0 + m] = silu_f(acc1[rr] + b1);
    }
    __syncthreads();

    // GEMM2: [32,128] x [128,128]-tile, no activation, + bias + residual
    v8f d0 = {0.f, 0.f, 0.f, 0.f, 0.f, 0.f, 0.f, 0.f};
    v8f d1 = {0.f, 0.f, 0.f, 0.f, 0.f, 0.f, 0.f, 0.f};
    {
      const float* arow = Ns + m * E_LD + h2;
      const v2f* bp = Bs2 + wv * (KS2 * 32) + lane;
#pragma unroll
      for (int ks = 0; ks < KS2; ++ks) {
        v2f b = bp[ks * 32];
        v2f a0 = *(const v2f*)(arow + 4 * ks);
        v2f a1 = *(const v2f*)(arow + 16 * E_LD + 4 * ks);
        d0 = wmma_k4(a0, b, d0);
        d1 = wmma_k4(a1, b, d1);
      }
    }
#pragma unroll
    for (int rr = 0; rr < 8; ++rr) {
      const int M0 = rr + 8 * hf;
#pragma unroll
      for (int half = 0; half < 2; ++half) {
        const int M = M0 + 16 * half;
        const long g = base + M;
        if (g < N) {
          float v = half ? d1[rr] : d0[rr];
          h_out[g * D + n0 + m] = h[g * D + n0 + m] + v + b2;
        }
      }
    }
  }
}

extern "C" void kernel_launch(void* const* d_in, const int* in_sizes, int n_in,
                              void* d_out, int out_size, void* d_ws,
                              size_t ws_size, hipStream_t stream) {
  const float* h = (const float*)d_in[0];
  const int* ei = (const int*)d_in[1];
  const float* coord = (const float*)d_in[2];
  // d_in[3] = num_nodes (scalar) — derived from in_sizes instead
  const float* l = (const float*)d_in[4];
  const float* We1 = (const float*)d_in[5];
  const float* be1 = (const float*)d_in[6];
  const float* We2 = (const float*)d_in[7];
  const float* be2 = (const float*)d_in[8];
  const float* Wn1 = (const float*)d_in[9];
  const float* bn1 = (const float*)d_in[10];
  const float* Wn2 = (const float*)d_in[11];
  const float* bn2 = (const float*)d_in[12];
  // d_in[13..15] = Wc1, bc1, Wc2: dead code in the reference

  const int N = in_sizes[0] / D;  // 50000
  const int E = in_sizes[1] / 2;  // 600000

  float* h_out = (float*)d_out;
  float* coord_out = h_out + (size_t)N * D;
  float* edge_feat = coord_out + (size_t)N * 3;
  float* agg = (float*)d_ws;  // [N, D] fp32 scratch for segment_sum

  {
    long n = (long)N * D;
    zero_f32<<<dim3((unsigned)((n + 255) / 256)), dim3(256), 0, stream>>>(agg,
                                                                          n);
  }

  // Persistent blocks: weights staged to LDS once per block, then grid-stride
  // over 32-row tiles.
  {
    long ntiles = ((long)E + TM - 1) / TM;
    unsigned nblk = (unsigned)((ntiles < 2048) ? ntiles : 2048);
    egcl_edge_kernel<<<dim3(nblk), dim3(256), 0, stream>>>(
        h, ei, coord, l, We1, be1, We2, be2, edge_feat, agg, E);
  }

  {
    int n3 = N * 3;
    coord_mod_kernel<<<dim3((unsigned)((n3 + 255) / 256)), dim3(256), 0,
                       stream>>>(coord, l, coord_out, n3);
  }

  {
    long ntiles = ((long)N + TM - 1) / TM;
    unsigned nblk = (unsigned)((ntiles < 1024) ? ntiles : 1024);
    egcl_node_kernel<<<dim3(nblk), dim3(256), 0, stream>>>(h, agg, Wn1, bn1,
                                                           Wn2, bn2, h_out, N);
  }
}